// GraphSRL_36034775614022
// MI455X (gfx1250) — compile-verified
//
#include <hip/hip_runtime.h>

typedef __attribute__((ext_vector_type(2))) float v2f;
typedef __attribute__((ext_vector_type(8))) float v8f;

#define Nn  8192
#define Ee  131072
#define FIN 512
#define DT  128
#define DL  16

// ---------------- utility kernels ----------------
__global__ void k_zero(float* p) {
    p[blockIdx.x * blockDim.x + threadIdx.x] = 0.f;
}

__global__ void k_fill_bias(float* out, const float* __restrict__ b, int D) {
    int t = blockIdx.x * blockDim.x + threadIdx.x;
    out[t] = b[t % D];
}

__global__ void k_degree(const long long* __restrict__ dst,
                         const float* __restrict__ ew, float* deg) {
    int e = blockIdx.x * blockDim.x + threadIdx.x;
    atomicAdd(&deg[(int)dst[e]], ew[e]);
}

__global__ void k_dinv(float* deg) {
    int i = blockIdx.x * blockDim.x + threadIdx.x;
    float d = deg[i] + 1.0f;                 // +1 = self-loop weight
    deg[i] = d > 0.f ? rsqrtf(d) : 0.f;
}

// ------------- fp32 WMMA GEMM: C[MxN] = A[MxK] @ B[KxN], one wave per 16x16 tile -------------
__global__ void k_gemm_wmma(const float* __restrict__ A, const float* __restrict__ B,
                            float* __restrict__ C, int K, int Ncols) {
    int wid  = blockIdx.x * (blockDim.x >> 5) + (threadIdx.x >> 5);
    int lane = threadIdx.x & 31;
    int half = lane >> 4;        // selects K-pair within K4 step
    int l16  = lane & 15;        // A row / B col within tile
    int tilesN = Ncols >> 4;
    int i0 = (wid / tilesN) << 4;
    int j0 = (wid % tilesN) << 4;
    int colB = j0 + l16;
    const float* Arow = A + (size_t)(i0 + l16) * K;
    v8f acc = {};
    for (int k0 = 0; k0 < K; k0 += 4) {
        int ka = k0 + (half << 1);
        v2f a = *(const v2f*)(Arow + ka);                 // A[l16][ka..ka+1]
        v2f b;
        b.x = B[(size_t)ka * Ncols + colB];               // B[ka][colB]
        b.y = B[(size_t)(ka + 1) * Ncols + colB];
        acc = __builtin_amdgcn_wmma_f32_16x16x4_f32(false, a, false, b,
                                                    (short)0, acc, false, false);
    }
    int rbase = i0 + (half << 3);
#pragma unroll
    for (int r = 0; r < 8; ++r)
        C[(size_t)(rbase + r) * Ncols + colB] = acc[r];
}

// ------------- GCN gather-scatter: out[dst] += h[src] * dinv[src]*w*dinv[dst] -------------
__global__ void k_scatter(const long long* __restrict__ src, const long long* __restrict__ dst,
                          const float* __restrict__ ew, const float* __restrict__ dinv,
                          const float* __restrict__ h, float* out, int D, int chunks) {
    int t = blockIdx.x * blockDim.x + threadIdx.x;
    int e = t / chunks;
    int c = t % chunks;
    int s, d; float w;
    if (e < Ee) { s = (int)src[e]; d = (int)dst[e]; w = ew[e]; }
    else        { s = d = e - Ee; w = 1.0f; }        // self-loop
    float norm = dinv[s] * w * dinv[d];
    const float4 hv = *(const float4*)(h + (size_t)s * D + c * 4);
    float* o = out + (size_t)d * D + c * 4;
    atomicAdd(o + 0, hv.x * norm);
    atomicAdd(o + 1, hv.y * norm);
    atomicAdd(o + 2, hv.z * norm);
    atomicAdd(o + 3, hv.w * norm);
}

// ------------- readout: out16[j] = relu(flat(xl) . Wro[:,j] + bro[j]) -------------
__global__ void k_readout(const float* __restrict__ xl, const float* __restrict__ Wro,
                          const float* __restrict__ bro, float* out16) {
    __shared__ float red[256];
    int j = blockIdx.x;                      // 0..15
    float acc = 0.f;
    for (int i = threadIdx.x; i < Nn * DL; i += blockDim.x)
        acc += xl[i] * Wro[(size_t)i * DL + j];
    red[threadIdx.x] = acc;
    __syncthreads();
    for (int s2 = 128; s2 > 0; s2 >>= 1) {
        if (threadIdx.x < s2) red[threadIdx.x] += red[threadIdx.x + s2];
        __syncthreads();
    }
    if (threadIdx.x == 0) out16[j] = fmaxf(red[0] + bro[j], 0.f);
}

__global__ void k_rownorm(const float* __restrict__ xl, float* nrm) {
    int i = blockIdx.x * blockDim.x + threadIdx.x;
    const float4* p = (const float4*)(xl + (size_t)i * DL);
    float s = 0.f;
#pragma unroll
    for (int c = 0; c < 4; ++c) {
        float4 v = p[c];
        s += v.x * v.x + v.y * v.y + v.z * v.z + v.w * v.w;
    }
    nrm[i] = s;
}

// ------------- decode: sim = 1/(1 + max(0, |xi|^2 + |xj|^2 - 2 xi.xj)); wave per 16x16 tile
// Output is a 537MB write-once stream (> 192MB L2) -> non-temporal stores. -------------
__global__ void k_decode(const float* __restrict__ xl, const float* __restrict__ nrm,
                         float* __restrict__ out) {
    int wid  = blockIdx.x * (blockDim.x >> 5) + (threadIdx.x >> 5);
    int lane = threadIdx.x & 31;
    int half = lane >> 4, l16 = lane & 15;
    int i0 = (wid >> 9) << 4;                 // 512 tiles per dim
    int j0 = (wid & 511) << 4;
    const float* arow = xl + (size_t)(i0 + l16) * DL;
    const float* brow = xl + (size_t)(j0 + l16) * DL;   // B = XL^T -> contiguous in XL
    v8f acc = {};
#pragma unroll
    for (int t = 0; t < 4; ++t) {
        int ka = (t << 2) + (half << 1);
        v2f a = *(const v2f*)(arow + ka);
        v2f b = *(const v2f*)(brow + ka);
        acc = __builtin_amdgcn_wmma_f32_16x16x4_f32(false, a, false, b,
                                                    (short)0, acc, false, false);
    }
    float nj = nrm[j0 + l16];
    int rbase = i0 + (half << 3);
#pragma unroll
    for (int r = 0; r < 8; ++r) {
        float sq = nrm[rbase + r] + nj - 2.f * acc[r];
        sq = fmaxf(sq, 0.f);
        __builtin_nontemporal_store(1.f / (1.f + sq),
                                    out + (size_t)(rbase + r) * Nn + (j0 + l16));
    }
}

extern "C" void kernel_launch(void* const* d_in, const int* in_sizes, int n_in,
                              void* d_out, int out_size, void* d_ws, size_t ws_size,
                              hipStream_t stream) {
    const float*     x[2]  = {(const float*)d_in[0],  (const float*)d_in[3]};
    const long long* ei[2] = {(const long long*)d_in[1], (const long long*)d_in[4]};
    const float*     ew[2] = {(const float*)d_in[2],  (const float*)d_in[5]};
    const float*     W1[2] = {(const float*)d_in[6],  (const float*)d_in[10]};
    const float*     b1[2] = {(const float*)d_in[7],  (const float*)d_in[11]};
    const float*     W2[2] = {(const float*)d_in[8],  (const float*)d_in[12]};
    const float*     b2[2] = {(const float*)d_in[9],  (const float*)d_in[13]};
    const float*     Wro   = (const float*)d_in[14];
    const float*     bro   = (const float*)d_in[15];

    float* out = (float*)d_out;
    float* xlat[2] = {out, out + (size_t)Nn * DL};
    float* summ[2] = {out + 2ull * Nn * DL, out + 2ull * Nn * DL + DL};
    float* dec[2]  = {out + 2ull * Nn * DL + 2 * DL,
                      out + 2ull * Nn * DL + 2 * DL + (size_t)Nn * Nn};

    float* ws   = (float*)d_ws;
    float* dinv = ws;                        // N floats (raw deg, then rsqrt in place)
    float* h1   = dinv + Nn;                 // N*DT
    float* agg1 = h1 + (size_t)Nn * DT;      // N*DT
    float* h2   = agg1 + (size_t)Nn * DT;    // N*DL
    float* nrm  = h2 + (size_t)Nn * DL;      // N

    const int BT = 256;                      // 8 waves/block (wave32)
    for (int g = 0; g < 2; ++g) {
        const long long* src = ei[g];
        const long long* dst = ei[g] + Ee;

        k_zero  <<<Nn / BT, BT, 0, stream>>>(dinv);
        k_degree<<<Ee / BT, BT, 0, stream>>>(dst, ew[g], dinv);
        k_dinv  <<<Nn / BT, BT, 0, stream>>>(dinv);

        // layer1: h1 = x @ W1  [8192x512]@[512x128]
        k_gemm_wmma<<<(Nn / 16) * (DT / 16) * 32 / BT, BT, 0, stream>>>(x[g], W1[g], h1, FIN, DT);
        k_fill_bias<<<Nn * DT / BT, BT, 0, stream>>>(agg1, b1[g], DT);
        k_scatter  <<<(Ee + Nn) * (DT / 4) / BT, BT, 0, stream>>>(src, dst, ew[g], dinv, h1, agg1, DT, DT / 4);

        // layer2: h2 = agg1 @ W2  [8192x128]@[128x16]
        k_gemm_wmma<<<(Nn / 16) * (DL / 16) * 32 / BT, BT, 0, stream>>>(agg1, W2[g], h2, DT, DL);
        k_fill_bias<<<Nn * DL / BT, BT, 0, stream>>>(xlat[g], b2[g], DL);
        k_scatter  <<<(Ee + Nn) * (DL / 4) / BT, BT, 0, stream>>>(src, dst, ew[g], dinv, h2, xlat[g], DL, DL / 4);

        // readout (deterministic per-column block reduction)
        k_readout<<<DL, BT, 0, stream>>>(xlat[g], Wro, bro, summ[g]);

        // pairwise similarity decode (bandwidth-bound NT store stream)
        k_rownorm<<<Nn / BT, BT, 0, stream>>>(xlat[g], nrm);
        k_decode <<<(Nn / 16) * (Nn / 16) * 32 / BT, BT, 0, stream>>>(xlat[g], nrm, dec[g]);
    }
}